// skeleton_80942953660575
// MI455X (gfx1250) — compile-verified
//
#include <hip/hip_runtime.h>

// ---------------- problem constants ----------------
#define IMG_H 1024
#define IMG_W 1024
#define NBATCH 16
#define NPIX (NBATCH * IMG_H * IMG_W)   // 16,777,216 floats = 64 MB per map
#define TILE 32
#define HALO 3
#define LDW (TILE + 2 * HALO)           // 38 rows loaded
#define LSTR 40                         // LDS row stride == TDM tile_dim0
#define SMOOTH 1e-5f
#define ITERS_GEO 10
#define ITERS_TOP 25
#define NBLK 2048                       // stage-1 reduction blocks

typedef __attribute__((ext_vector_type(2))) float v2f;
typedef __attribute__((ext_vector_type(8))) float v8f;
typedef __attribute__((ext_vector_type(4))) unsigned int v4u;
typedef __attribute__((ext_vector_type(4))) int v4i;
typedef __attribute__((ext_vector_type(8))) int v8i;

#define PINF __builtin_inff()

// ---------------------------------------------------------------------------
// Fused soft-skeleton step.
//  DO_ERODE=false (init):  img' = x;          skel  = relu(x  - dilate(erode(x)))
//  DO_ERODE=true  (iter):  img' = erode(img); delta = relu(img' - dilate(erode(img')))
//                          skel += relu(delta - skel*delta)
// blockIdx.z < NBATCH -> target image, else prediction image.
// Interior tiles are loaded by the Tensor Data Mover (one 2D-tile DMA into
// LDS, TENSORcnt-tracked); border tiles fall back to masked manual loads
// because TDM OOB returns 0 (we need +inf padding for the min-erosion).
// ---------------------------------------------------------------------------
template <bool DO_ERODE>
__global__ void __launch_bounds__(256)
skel_step(const float* __restrict__ src_t, const float* __restrict__ src_p,
          float* __restrict__ dst_t, float* __restrict__ dst_p,
          float* __restrict__ skl_t, float* __restrict__ skl_p) {
  __shared__ float s_img[LDW][LSTR];      // 38x40 input tile (halo 3)
  __shared__ float s_e1[LDW - 2][LSTR];   // 36x36 first erode (halo 2)
  __shared__ float s_e2[LDW - 4][LSTR];   // 34x34 second erode (halo 1)

  const int z = blockIdx.z;
  const bool is_t = (z < NBATCH);
  const int batch = is_t ? z : z - NBATCH;
  const float* __restrict__ src = is_t ? src_t : src_p;
  float* __restrict__ dst = is_t ? dst_t : dst_p;
  float* __restrict__ skl = is_t ? skl_t : skl_p;
  const size_t base = (size_t)batch * IMG_H * IMG_W;

  const int gx0 = blockIdx.x * TILE;
  const int gy0 = blockIdx.y * TILE;
  const int tid = threadIdx.x;

  // Interior: 40-wide x 38-tall loaded window fully inside this batch image.
  const bool interior = (gx0 >= HALO) && (gx0 - HALO + LSTR <= IMG_W) &&
                        (gy0 >= HALO) && (gy0 - HALO + LDW <= IMG_H);

  if (interior) {
    if (tid < 32) {   // one wave issues the DMA (TDM ignores EXEC)
      const float* gsrc = src + base + (size_t)(gy0 - HALO) * IMG_W + (gx0 - HALO);
      unsigned long long ga = (unsigned long long)(uintptr_t)gsrc;
      unsigned int lds_off =
          (unsigned int)(uintptr_t)(&s_img[0][0]);
      // D# group0: count=1 | lds_addr | global_addr[56:0] | type=2
      v4u g0;
      g0[0] = 1u;                                   // count=1, user mode
      g0[1] = lds_off;                              // LDS byte address
      g0[2] = (unsigned int)(ga & 0xffffffffu);     // global_addr[31:0]
      g0[3] = (unsigned int)((ga >> 32) & 0x01ffffffu) | (2u << 30); // [56:32]|type=2
      // D# group1: data_size=4B, tensor 1024x1024 stride 1024, tile 40x38
      v8i g1;
      g1[0] = 0x00020000;          // wgmask=0, data_size=2 (4 bytes)
      g1[1] = (int)(1024u << 16);  // tensor_dim0[15:0]=1024 (bits 63:48)
      g1[2] = (int)(1024u << 16);  // tensor_dim0 hi=0, tensor_dim1[15:0]=1024
      g1[3] = (int)((unsigned)LSTR << 16); // tensor_dim1 hi=0, tile_dim0=40
      g1[4] = LDW;                 // tile_dim1=38, tile_dim2=0
      g1[5] = 1024;                // tensor_dim0_stride[31:0]=1024
      g1[6] = 0;                   // stride hi, tensor_dim1_stride lo
      g1[7] = 0;
      v4i gz = {0, 0, 0, 0};
#if __clang_major__ >= 23
      v8i gz8 = {0, 0, 0, 0, 0, 0, 0, 0};
      __builtin_amdgcn_tensor_load_to_lds(g0, g1, gz, gz, gz8, 0);
#else
      __builtin_amdgcn_tensor_load_to_lds(g0, g1, gz, gz, 0);
#endif
      __builtin_amdgcn_s_wait_tensorcnt(0);
    }
  } else {
    // ---- masked manual load, out-of-image = +inf (identity for min) ----
    for (int i = tid; i < LDW * LDW; i += 256) {
      int ly = i / LDW, lx = i % LDW;
      int gy = gy0 + ly - HALO, gx = gx0 + lx - HALO;
      float v = PINF;
      if (gy >= 0 && gy < IMG_H && gx >= 0 && gx < IMG_W)
        v = src[base + (size_t)gy * IMG_W + gx];
      s_img[ly][lx] = v;
    }
  }
  __syncthreads();

  // ---- E1 (cross min or identity) on 36x36; force +inf outside image ----
  for (int i = tid; i < 36 * 36; i += 256) {
    int ly = i / 36, lx = i % 36;
    int gy = gy0 + ly - 2, gx = gx0 + lx - 2;
    float v = PINF;
    if (gy >= 0 && gy < IMG_H && gx >= 0 && gx < IMG_W) {
      float c = s_img[ly + 1][lx + 1];
      if (DO_ERODE) {
        float vm = fminf(fminf(s_img[ly][lx + 1], c), s_img[ly + 2][lx + 1]);
        float hm = fminf(fminf(s_img[ly + 1][lx], c), s_img[ly + 1][lx + 2]);
        v = fminf(vm, hm);
      } else {
        v = c;
      }
    }
    s_e1[ly][lx] = v;
  }
  __syncthreads();

  // ---- E2 = cross-min(E1) on 34x34; force +inf outside image ----
  for (int i = tid; i < 34 * 34; i += 256) {
    int ly = i / 34, lx = i % 34;
    int gy = gy0 + ly - 1, gx = gx0 + lx - 1;
    float v = PINF;
    if (gy >= 0 && gy < IMG_H && gx >= 0 && gx < IMG_W) {
      float c = s_e1[ly + 1][lx + 1];
      float vm = fminf(fminf(s_e1[ly][lx + 1], c), s_e1[ly + 2][lx + 1]);
      float hm = fminf(fminf(s_e1[ly + 1][lx], c), s_e1[ly + 1][lx + 2]);
      v = fminf(vm, hm);
    }
    s_e2[ly][lx] = v;
  }
  __syncthreads();

  // ---- open = 3x3 max of E2 (-inf image padding); delta; skel update ----
  // Each thread owns 4 consecutive pixels of one row -> b128 loads/stores.
  {
    const int ly = tid >> 3;             // 0..31
    const int lx0 = (tid & 7) * 4;       // 0,4,...,28
    const int gy = gy0 + ly;
    const size_t g0i = base + (size_t)gy * IMG_W + gx0 + lx0;

    float4 e1c, dlt;
#pragma unroll
    for (int j = 0; j < 4; ++j) {
      const int lx = lx0 + j;
      const int gx = gx0 + lx;
      float o = -PINF;
#pragma unroll
      for (int dy = -1; dy <= 1; ++dy)
#pragma unroll
        for (int dx = -1; dx <= 1; ++dx) {
          int ny = gy + dy, nx = gx + dx;
          if (ny >= 0 && ny < IMG_H && nx >= 0 && nx < IMG_W)
            o = fmaxf(o, s_e2[ly + 1 + dy][lx + 1 + dx]);
        }
      float e1 = s_e1[ly + 2][lx + 2];   // img' at this pixel
      (&e1c.x)[j] = e1;
      (&dlt.x)[j] = fmaxf(e1 - o, 0.0f);
    }

    float4 snew;
    if (DO_ERODE) {
      float4 s = *(const float4*)(skl + g0i);
#pragma unroll
      for (int j = 0; j < 4; ++j) {
        float sv = (&s.x)[j], dv = (&dlt.x)[j];
        (&snew.x)[j] = sv + fmaxf(dv - sv * dv, 0.0f);
      }
    } else {
      snew = dlt;
    }
    *(float4*)(dst + g0i) = e1c;
    *(float4*)(skl + g0i) = snew;
  }
}

// ---------------------------------------------------------------------------
// Stage-1 reduction (b128 streams): per-block partial sums of
//   q0=sum(skel_pred), q1=sum(skel_pred*y_true), q2=sum(skel_true), q3=sum(skel_true*y_pred)
// Block combine on the matrix unit: B = ones(4x16), A = 64 partials per call
// -> V_WMMA_F32_16X16X4_F32 chained 4x reduces 256 f32 values exactly.
// ---------------------------------------------------------------------------
__global__ void __launch_bounds__(256)
skel_reduce(const float* __restrict__ skl_t, const float* __restrict__ skl_p,
            const float* __restrict__ y_true, const float* __restrict__ y_pred,
            float* __restrict__ part) {
  __shared__ float red[4][256];
  const float4* st4 = (const float4*)skl_t;
  const float4* sp4 = (const float4*)skl_p;
  const float4* yt4 = (const float4*)y_true;
  const float4* yp4 = (const float4*)y_pred;

  float p0 = 0.f, p1 = 0.f, p2 = 0.f, p3 = 0.f;
  const int stride = gridDim.x * 256;
  for (int i = blockIdx.x * 256 + threadIdx.x; i < NPIX / 4; i += stride) {
    float4 sp = sp4[i], st = st4[i], yt = yt4[i], yp = yp4[i];
#pragma unroll
    for (int j = 0; j < 4; ++j) {
      float spv = (&sp.x)[j], stv = (&st.x)[j];
      p0 += spv;
      p1 += spv * (&yt.x)[j];
      p2 += stv;
      p3 += stv * (&yp.x)[j];
    }
  }
  red[0][threadIdx.x] = p0;
  red[1][threadIdx.x] = p1;
  red[2][threadIdx.x] = p2;
  red[3][threadIdx.x] = p3;
  __syncthreads();

  if (threadIdx.x < 32) {        // wave 0 only: EXEC all-ones for WMMA
    const int lane = threadIdx.x;
    v2f ones;
    ones[0] = 1.0f;
    ones[1] = 1.0f;
#pragma unroll
    for (int q = 0; q < 4; ++q) {
      v8f c = {};
#pragma unroll
      for (int chunk = 0; chunk < 4; ++chunk) {
        v2f a;
        a[0] = red[q][chunk * 64 + 2 * lane];
        a[1] = red[q][chunk * 64 + 2 * lane + 1];
        c = __builtin_amdgcn_wmma_f32_16x16x4_f32(
            false, a, false, ones, (short)0, c, false, false);
      }
      float t = c[0] + c[1] + c[2] + c[3] + c[4] + c[5] + c[6] + c[7];
      int sw = __builtin_amdgcn_ds_swizzle(__float_as_int(t), 0x401f); // SWAPX16
      float tot = t + __int_as_float(sw);
      if (lane == 0) part[q * NBLK + blockIdx.x] = tot;
    }
  }
}

// Stage-2: deterministic tree reduction of NBLK partials per quantity.
__global__ void __launch_bounds__(256)
reduce_final(const float* __restrict__ part, float* __restrict__ acc) {
  __shared__ float red[256];
  const int q = blockIdx.x;   // 4 blocks
  float s = 0.f;
  for (int i = threadIdx.x; i < NBLK; i += 256) s += part[q * NBLK + i];
  red[threadIdx.x] = s;
  __syncthreads();
  for (int w = 128; w > 0; w >>= 1) {
    if (threadIdx.x < w) red[threadIdx.x] += red[threadIdx.x + w];
    __syncthreads();
  }
  if (threadIdx.x == 0) acc[q] = red[0];
}

__global__ void finalize_kernel(const float* __restrict__ acc, float* __restrict__ out) {
  if (threadIdx.x == 0) {
#pragma unroll
    for (int j = 0; j < 2; ++j) {
      const float* a = acc + 4 * j;
      float tprec = (a[1] + SMOOTH) / (a[0] + SMOOTH);
      float tsens = (a[3] + SMOOTH) / (a[2] + SMOOTH);
      out[j] = 1.0f - 2.0f * (tprec * tsens) / (tprec + tsens);
    }
  }
}

// ---------------------------------------------------------------------------
extern "C" void kernel_launch(void* const* d_in, const int* in_sizes, int n_in,
                              void* d_out, int out_size, void* d_ws, size_t ws_size,
                              hipStream_t stream) {
  const float* target = (const float*)d_in[0];  // y_true (binary mask)
  const float* inputs = (const float*)d_in[1];  // y_pred (probabilities)
  float* out = (float*)d_out;                   // [cl_dice_geo, cl_dice_top]

  float* base = (float*)d_ws;
  float* acc  = base;               // 8 floats (geo: 0..3, top: 4..7)
  float* part = base + 16;          // 4*NBLK partial sums
  float* big  = base + 16384;       // 6 x 64MB ping-pong/state buffers
  float* imgA_t = big;
  float* imgB_t = imgA_t + NPIX;
  float* skl_t  = imgB_t + NPIX;
  float* imgA_p = skl_t + NPIX;
  float* imgB_p = imgA_p + NPIX;
  float* skl_p  = imgB_p + NPIX;

  dim3 grid(IMG_W / TILE, IMG_H / TILE, 2 * NBATCH);
  dim3 blk(256);

  // init: img = x, skel = relu(x - open(x))  (both images in one launch)
  skel_step<false><<<grid, blk, 0, stream>>>(target, inputs, imgA_t, imgA_p, skl_t, skl_p);

  float* st = imgA_t; float* dt = imgB_t;
  float* sp = imgA_p; float* dp = imgB_p;
  for (int it = 1; it <= ITERS_TOP; ++it) {
    skel_step<true><<<grid, blk, 0, stream>>>(st, sp, dt, dp, skl_t, skl_p);
    float* tmp;
    tmp = st; st = dt; dt = tmp;
    tmp = sp; sp = dp; dp = tmp;
    if (it == ITERS_GEO) {          // snapshot reductions for the 10-iter skeleton
      skel_reduce<<<NBLK, blk, 0, stream>>>(skl_t, skl_p, target, inputs, part);
      reduce_final<<<4, blk, 0, stream>>>(part, acc + 0);
    }
  }
  skel_reduce<<<NBLK, blk, 0, stream>>>(skl_t, skl_p, target, inputs, part);
  reduce_final<<<4, blk, 0, stream>>>(part, acc + 4);
  finalize_kernel<<<1, 32, 0, stream>>>(acc, out);
}